// _YOLOTargetLayer_86878598463653
// MI455X (gfx1250) — compile-verified
//
#include <hip/hip_runtime.h>
#include <stdint.h>

// ---------------- problem constants ----------------
#define B_    32
#define HW_   1600
#define A_    5
#define N_    128
#define C_    80
#define WOUT_ 40.0f
#define HOUT_ 40.0f
#define WIN_  1280.0f
#define HIN_  1280.0f
#define STRIDE_ 32.0f
#define IOU_THRESH_ 0.6f

static constexpr int ITEMS = HW_ * A_;              // 8000 per batch
static constexpr int TOT   = B_ * ITEMS;            // 256000
// flat float offsets of the 6 concatenated outputs (reference return order)
static constexpr int OFF_BOXES = 0;                 // (B,HW,A,4)
static constexpr int OFF_IOUS  = TOT * 4;           // 1,024,000
static constexpr int OFF_CLS   = OFF_IOUS + TOT;    // 1,280,000
static constexpr int OFF_BM    = OFF_CLS + TOT*C_;  // 21,760,000
static constexpr int OFF_IM    = OFF_BM + TOT;      // 22,016,000
static constexpr int OFF_CM    = OFF_IM + TOT;      // 22,272,000
static constexpr int OUT_TOTAL = OFF_CM + TOT;      // 22,528,000 floats

typedef __attribute__((ext_vector_type(4))) float        v4f;
typedef __attribute__((ext_vector_type(4))) unsigned int v4u;
typedef __attribute__((ext_vector_type(8))) unsigned int v8u;

// ---------------- kernel 1: streaming default fill (bandwidth-bound, ~90MB) ----
__global__ __launch_bounds__(256) void fill_defaults(v4f* __restrict__ out)
{
    int i = blockIdx.x * 256 + threadIdx.x;          // float4 index
    if (i >= OUT_TOTAL / 4) return;
    v4f v;
    if (i < OFF_IOUS / 4) {
        v.x = 0.5f; v.y = 0.5f; v.z = 1.0f; v.w = 1.0f;     // boxes_t default
    } else if (i >= OFF_BM / 4 && i < OFF_IM / 4) {
        v.x = v.y = v.z = v.w = 0.01f;                       // box_mask default
    } else {
        v.x = v.y = v.z = v.w = 0.0f;                        // ious/classes/iou_mask/class_mask
    }
    __builtin_nontemporal_store(v, &out[i]);                 // NT: don't pollute L2 hot set
}

// ---------------- shared box-decode (must match reference op order exactly) ---
__device__ __forceinline__ void pred_box(float4 bp, int hw, float ax, float ay,
                                         float& x1, float& y1, float& x2, float& y2)
{
    float col = (float)(hw % 40);
    float row = (float)(hw / 40);
    float cxp = (bp.x + col) / WOUT_;
    float cyp = (bp.y + row) / HOUT_;
    float bw  = bp.z * ax / WOUT_ * 0.5f;
    float bh  = bp.w * ay / HOUT_ * 0.5f;
    x1 = (cxp - bw) * WIN_;  y1 = (cyp - bh) * HIN_;
    x2 = (cxp + bw) * WIN_;  y2 = (cyp + bh) * HIN_;
}

// ---------------- kernel 2: best-IoU over GT -> iou_mask ----------------------
// One TDM tensor_load_to_lds per block stages gt_boxes[b] (640 dwords) into LDS.
__global__ __launch_bounds__(256)
void yolo_main(const float* __restrict__ bbox_pred, const float* __restrict__ iou_pred,
               const float* __restrict__ gt, const float* __restrict__ anchors,
               float* __restrict__ out)
{
    __shared__ float sgt[N_ * 5];
    __shared__ float sgarea[N_];
    const int b   = blockIdx.y;
    const int tid = threadIdx.x;

    if (tid < 32) {   // wave 0 only issues the TDM DMA (EXEC ignored by TDM)
        // D# group 0 (ISA 8.3): count=1 | lds_addr | global_addr | type=2
        uint32_t lds = (uint32_t)(uintptr_t)(&sgt[0]);
        uint64_t ga  = (uint64_t)(uintptr_t)(gt + (size_t)b * (N_ * 5));
        v4u g0;
        g0.s0 = 1u;                          // count=1, user descriptor
        g0.s1 = lds;                         // lds_addr
        g0.s2 = (uint32_t)ga;                // global_addr[31:0]
        g0.s3 = (uint32_t)((ga >> 32) & 0x01FFFFFFu) | (2u << 30);  // ga[56:32] | type=2
        // D# group 1 (ISA 8.4): 1-row tile of 640 4-byte elements
        v8u g1;
        g1.s0 = 0x00020000u;                 // wg_mask=0, data_size=2 (4B)
        g1.s1 = (640u & 0xFFFFu) << 16;      // tensor_dim0[15:0] @ bits[63:48]
        g1.s2 = (640u >> 16) | (1u << 16);   // tensor_dim0[31:16] | tensor_dim1=1
        g1.s3 = (640u << 16);                // tile_dim0=640 @ bits[127:112]
        g1.s4 = 1u;                          // tile_dim1=1, tile_dim2=0
        g1.s5 = 640u;                        // tensor_dim0_stride[31:0]
        g1.s6 = 0u;                          // stride0[47:32]=0, stride1 lo=0
        g1.s7 = 0u;
        asm volatile("tensor_load_to_lds %0, %1" :: "s"(g0), "s"(g1) : "memory");
        __builtin_amdgcn_s_wait_tensorcnt(0);
    }
    __syncthreads();

    if (tid < N_) {
        const float* g = &sgt[tid * 5];
        sgarea[tid] = (g[2] - g[0] + 1.0f) * (g[3] - g[1] + 1.0f);
    }
    __syncthreads();

    int item = blockIdx.x * 256 + tid;       // item = hw*A + a
    if (item >= ITEMS) return;
    int hw = item / 5;
    int a  = item - hw * 5;

    float4 bp = ((const float4*)bbox_pred)[(size_t)b * ITEMS + item];
    float ax = anchors[2 * a], ay = anchors[2 * a + 1];
    float x1, y1, x2, y2;
    pred_box(bp, hw, ax, ay, x1, y1, x2, y2);
    float barea = (x2 - x1 + 1.0f) * (y2 - y1 + 1.0f);

    float best = 0.0f;                       // IoUs are >= 0
    #pragma unroll 4
    for (int n = 0; n < N_; ++n) {
        float gx1 = sgt[n*5+0], gy1 = sgt[n*5+1], gx2 = sgt[n*5+2], gy2 = sgt[n*5+3];
        float iw = fmaxf(fminf(x2, gx2) - fmaxf(x1, gx1) + 1.0f, 0.0f);
        float ih = fmaxf(fminf(y2, gy2) - fmaxf(y1, gy1) + 1.0f, 0.0f);
        float inter = iw * ih;
        float iou = inter / (barea + sgarea[n] - inter);
        best = fmaxf(best, iou);
    }
    int flat = b * ITEMS + item;
    out[OFF_IM + flat] = (best < IOU_THRESH_) ? -iou_pred[flat] : 0.0f;
}

// ---------------- kernel 3: per-GT scatter (last-write-wins, n-ordered) -------
__global__ __launch_bounds__(128)
void yolo_scatter(const float* __restrict__ bbox_pred, const float* __restrict__ iou_pred,
                  const float* __restrict__ gt, const float* __restrict__ anchors,
                  const int* __restrict__ num_boxes, float* __restrict__ out)
{
    __shared__ float  s_g[N_][5];
    __shared__ float4 s_tb[N_];
    __shared__ int    s_flat[N_], s_cls[N_], s_valid[N_];
    __shared__ float  s_anc[2 * A_];

    const int b = blockIdx.x;
    const int n = threadIdx.x;

    if (n < 2 * A_) s_anc[n] = anchors[n];
    const float* grow = gt + ((size_t)b * N_ + n) * 5;
    float g0 = grow[0], g1 = grow[1], g2 = grow[2], g3 = grow[3], g4 = grow[4];
    s_g[n][0] = g0; s_g[n][1] = g1; s_g[n][2] = g2; s_g[n][3] = g3; s_g[n][4] = g4;
    __syncthreads();

    int nb = num_boxes[b];
    float cx  = (g0 + g2) * 0.5f / STRIDE_;
    float cy  = (g1 + g3) * 0.5f / STRIDE_;
    float fcx = floorf(cx), fcy = floorf(cy);
    int   cell = (int)(fcy * WOUT_ + fcx);
    float tw = (g2 - g0 + 1.0f) / STRIDE_;
    float th = (g3 - g1 + 1.0f) / STRIDE_;
    float gw = g2 / STRIDE_ - g0 / STRIDE_ + 1.0f;   // matches gr[...,2]-gr[...,0]+1
    float gh = g3 / STRIDE_ - g1 / STRIDE_ + 1.0f;

    float bestv = -1.0f; int bi = 0;                 // argmax -> first occurrence
    for (int a = 0; a < A_; ++a) {
        float aw = s_anc[2*a], ah = s_anc[2*a+1];
        float iwa = fminf(aw, gw), iha = fminf(ah, gh);
        float inta = iwa * iha;
        float aiou = inta / (aw * ah + gw * gh - inta);
        if (aiou > bestv) { bestv = aiou; bi = a; }
    }

    float4 tb;
    tb.x = cx - fcx;  tb.y = cy - fcy;
    tb.z = tw / s_anc[2*bi];  tb.w = th / s_anc[2*bi+1];
    s_tb[n]    = tb;
    s_flat[n]  = (b * HW_ + cell) * A_ + bi;
    s_cls[n]   = (int)g4;
    s_valid[n] = (n < nb) && (cell >= 0) && (cell < HW_);
    __syncthreads();

    if (threadIdx.x == 0) {
        // serial in n => deterministic last-write-wins on duplicate flat indices,
        // matching sequential scatter semantics of the reference
        for (int k = 0; k < N_; ++k) {
            if (!s_valid[k]) continue;
            int f = s_flat[k];
            int r = f - b * ITEMS;
            int cell2 = r / 5;
            int a2    = r - cell2 * 5;

            out[OFF_IM + f] = 5.0f * (1.0f - iou_pred[f]);

            float4 bp = ((const float4*)bbox_pred)[f];
            float x1, y1, x2, y2;
            pred_box(bp, cell2, s_anc[2*a2], s_anc[2*a2+1], x1, y1, x2, y2);
            float barea = (x2 - x1 + 1.0f) * (y2 - y1 + 1.0f);
            float gx1 = s_g[k][0], gy1 = s_g[k][1], gx2 = s_g[k][2], gy2 = s_g[k][3];
            float garea = (gx2 - gx1 + 1.0f) * (gy2 - gy1 + 1.0f);
            float iw = fmaxf(fminf(x2, gx2) - fmaxf(x1, gx1) + 1.0f, 0.0f);
            float ih = fmaxf(fminf(y2, gy2) - fmaxf(y1, gy1) + 1.0f, 0.0f);
            float inter = iw * ih;
            out[OFF_IOUS + f] = inter / (barea + garea - inter);

            out[OFF_BM + f] = 1.0f;
            ((float4*)(out + OFF_BOXES))[f] = s_tb[k];
            out[OFF_CM + f] = 1.0f;
            out[OFF_CLS + (size_t)f * C_ + s_cls[k]] = 1.0f;
        }
    }
}

// ---------------- launcher ----------------------------------------------------
extern "C" void kernel_launch(void* const* d_in, const int* in_sizes, int n_in,
                              void* d_out, int out_size, void* d_ws, size_t ws_size,
                              hipStream_t stream)
{
    (void)in_sizes; (void)n_in; (void)out_size; (void)d_ws; (void)ws_size;
    const float* bbox = (const float*)d_in[0];
    const float* ioup = (const float*)d_in[1];
    const float* gtb  = (const float*)d_in[2];
    const float* anch = (const float*)d_in[3];
    const int*   nb   = (const int*)d_in[4];
    float* out = (float*)d_out;

    fill_defaults<<<dim3(OUT_TOTAL / 4 / 256), dim3(256), 0, stream>>>((v4f*)out);
    yolo_main<<<dim3((ITEMS + 255) / 256, B_), dim3(256), 0, stream>>>(bbox, ioup, gtb, anch, out);
    yolo_scatter<<<dim3(B_), dim3(N_), 0, stream>>>(bbox, ioup, gtb, anch, nb, out);
}